// S6_24008867184849
// MI455X (gfx1250) — compile-verified
//
#include <hip/hip_runtime.h>
#include <hip/hip_bf16.h>
#include <math.h>

// S6 (Mamba-style) forward with h0 = 0 collapses to:
//   y[t,d] = x[t,d] * softplus(xW1 + b1)[t,d] * ( x M x^T + u.x + c )[t]
// where M = W2 W3^T (16x16), u = W2 b3 + W3 b2, c = b2.b3.
// Everything is done with f32 WMMA (V_WMMA_F32_16X16X4_F32) in registers.

typedef __attribute__((ext_vector_type(2))) float v2f;   // f32 WMMA A/B fragment (2 VGPRs)
typedef __attribute__((ext_vector_type(8))) float v8f;   // f32 WMMA C/D fragment (8 VGPRs)

#define D_DIM 16
#define N_DIM 1024

__device__ __forceinline__ float softplus_f(float v) {
    // numerically stable log1p(exp(v))
    return fmaxf(v, 0.0f) + log1pf(expf(-fabsf(v)));
}

// ---------------------------------------------------------------------------
// Kernel 1 (one wave): Mred = W2 @ W3^T via 256 chained WMMA f32 16x16x4,
// plus u[j] = W2[j,:].b3 + W3[j,:].b2 and c = b2.b3.
// A-frag: lane (hi,lo): a = {W2[lo][4k+2hi], W2[lo][4k+2hi+1]}  (contiguous)
// B-frag (W3^T): b = {W3[lo][4k+2hi], W3[lo][4k+2hi+1]}         (same addressing)
// ---------------------------------------------------------------------------
__global__ __launch_bounds__(32) void s6_precompute_kernel(
    const float* __restrict__ W2, const float* __restrict__ b2,
    const float* __restrict__ W3, const float* __restrict__ b3,
    float* __restrict__ wsM, float* __restrict__ wsU, float* __restrict__ wsC)
{
    const int lane = threadIdx.x;      // exactly one full wave32: EXEC all 1s
    const int hi = lane >> 4;
    const int lo = lane & 15;

    v8f acc = {};
    for (int k = 0; k < N_DIM / 4; ++k) {
        const int col = 4 * k + 2 * hi;
        v2f a = *(const v2f*)(W2 + lo * N_DIM + col);
        v2f b = *(const v2f*)(W3 + lo * N_DIM + col);
        acc = __builtin_amdgcn_wmma_f32_16x16x4_f32(false, a, false, b,
                                                    (short)0, acc, false, false);
    }
    // C/D layout: VGPR r holds row (r + 8*hi), column lo -> row-major store
    #pragma unroll
    for (int r = 0; r < 8; ++r)
        wsM[(r + 8 * hi) * D_DIM + lo] = acc[r];

    if (lane < 16) {
        float s = 0.f;
        for (int n = 0; n < N_DIM; ++n)
            s += W2[lane * N_DIM + n] * b3[n] + W3[lane * N_DIM + n] * b2[n];
        wsU[lane] = s;
    }
    if (lane == 16) {
        float s = 0.f;
        for (int n = 0; n < N_DIM; ++n) s += b2[n] * b3[n];
        wsC[0] = s;
    }
}

// ---------------------------------------------------------------------------
// Kernel 2: one wave per 16-token tile (400 tiles = 6400 tokens).
//   T = X @ Mred, P = X @ W1 (two 16x16x16 f32 GEMMs, 4 WMMA steps each)
//   s_i = sum_j (T[i,j] + u[j]) * X[i,j] + c   (16-lane shfl_xor reduction)
//   y[i,j] = X[i,j] * softplus(P[i,j] + b1[j]) * s_i
// ---------------------------------------------------------------------------
__global__ __launch_bounds__(256) void s6_token_kernel(
    const float* __restrict__ x, const float* __restrict__ W1,
    const float* __restrict__ b1,
    const float* __restrict__ wsM, const float* __restrict__ wsU,
    const float* __restrict__ wsC,
    float* __restrict__ y, int n_tiles)
{
    const int wave = (int)((blockIdx.x * blockDim.x + threadIdx.x) >> 5);
    if (wave >= n_tiles) return;            // uniform per wave: EXEC stays all-1s
    const int lane = threadIdx.x & 31;
    const int hi = lane >> 4;
    const int lo = lane & 15;

    const float* __restrict__ xt = x + wave * (16 * D_DIM);
    float* __restrict__ yt = y + wave * (16 * D_DIM);

    // A fragments of the 16(token) x 16(dim) tile X: contiguous float2 loads
    v2f a[4];
    #pragma unroll
    for (int k = 0; k < 4; ++k)
        a[k] = *(const v2f*)(xt + lo * D_DIM + 4 * k + 2 * hi);

    v8f T = {};   // X @ Mred
    v8f P = {};   // X @ W1
    #pragma unroll
    for (int k = 0; k < 4; ++k) {
        const int r0 = 4 * k + 2 * hi;
        v2f bM = { wsM[r0 * D_DIM + lo], wsM[(r0 + 1) * D_DIM + lo] };
        v2f bW = { W1[r0 * D_DIM + lo],  W1[(r0 + 1) * D_DIM + lo]  };
        T = __builtin_amdgcn_wmma_f32_16x16x4_f32(false, a[k], false, bM,
                                                  (short)0, T, false, false);
        P = __builtin_amdgcn_wmma_f32_16x16x4_f32(false, a[k], false, bW,
                                                  (short)0, P, false, false);
    }

    const float uv  = wsU[lo];
    const float b1v = b1[lo];
    const float c0  = wsC[0];

    #pragma unroll
    for (int r = 0; r < 8; ++r) {
        const int row = r + 8 * hi;
        const float xc = xt[row * D_DIM + lo];        // X in C/D layout
        float t = (T[r] + uv) * xc;
        // reduce across the 16 lanes that hold this row's columns
        t += __shfl_xor(t, 1, 16);
        t += __shfl_xor(t, 2, 16);
        t += __shfl_xor(t, 4, 16);
        t += __shfl_xor(t, 8, 16);
        const float s = t + c0;
        const float delta = softplus_f(P[r] + b1v);
        yt[row * D_DIM + lo] = xc * delta * s;
    }
}

extern "C" void kernel_launch(void* const* d_in, const int* in_sizes, int n_in,
                              void* d_out, int out_size, void* d_ws, size_t ws_size,
                              hipStream_t stream) {
    const float* x  = (const float*)d_in[0];
    const float* W1 = (const float*)d_in[1];
    const float* b1 = (const float*)d_in[2];
    const float* W2 = (const float*)d_in[3];
    const float* b2 = (const float*)d_in[4];
    const float* W3 = (const float*)d_in[5];
    const float* b3 = (const float*)d_in[6];
    // d_in[7] = A_log: mathematically dead (h0 == 0 => dA * h == 0)

    float* wsM = (float*)d_ws;       // 256 f32
    float* wsU = wsM + 256;          // 16  f32
    float* wsC = wsU + 16;           // 1   f32

    s6_precompute_kernel<<<1, 32, 0, stream>>>(W2, b2, W3, b3, wsM, wsU, wsC);

    const int tokens  = in_sizes[0] / D_DIM;   // 6400
    const int n_tiles = tokens / 16;           // 400
    const int waves_per_block = 8;             // 256 threads = 8 wave32
    const int blocks = (n_tiles + waves_per_block - 1) / waves_per_block;

    s6_token_kernel<<<blocks, 256, 0, stream>>>(x, W1, b1, wsM, wsU, wsC,
                                                (float*)d_out, n_tiles);
}